// VoronoiPropagation_16372415333125
// MI455X (gfx1250) — compile-verified
//
#include <hip/hip_runtime.h>
#include <hip/hip_bf16.h>

typedef __attribute__((ext_vector_type(2))) float v2f;
typedef __attribute__((ext_vector_type(8))) float v8f;

namespace {
constexpr int kH = 224, kW = 224, kB = 64, kC = 196;
constexpr int kHW   = kH * kW;          // 50176
constexpr int kNPix = kB * kHW;         // 3,211,264
constexpr int kNBnd = 10;               // NB
constexpr int kWin  = 20;               // 2*NB
constexpr int kIters = 50;
constexpr float kColorW = 10.0f;
constexpr int kBand = 8;                // rows per sobel block
constexpr int kTileRows = kBand + 2;    // 10 (halo)
constexpr int kTileCols = kW + 2;       // 226 (halo)
}

__device__ __forceinline__ unsigned lds_addr_u32(const void* p) {
    // Generic pointers into the LDS aperture carry the LDS byte offset in
    // their low 32 bits (ISA 10.2).
    return (unsigned)(uintptr_t)p;
}

// ---------------------------------------------------------------------------
// Gray conversion (memory-bound streaming)
// ---------------------------------------------------------------------------
__global__ void k_gray(const float* __restrict__ x, float* __restrict__ gray) {
    int i = blockIdx.x * blockDim.x + threadIdx.x;
    if (i >= kNPix) return;
    int b = i / kHW, p = i % kHW;
    const float* xb = x + (size_t)b * 3 * kHW;
    gray[i] = 0.2989f * xb[p] + 0.587f * xb[kHW + p] + 0.114f * xb[2 * kHW + p];
}

// ---------------------------------------------------------------------------
// Sobel via WMMA micro-GEMM over an async-staged LDS tile.
//   Block: 8 waves == 8 image rows (one band), full 224-wide.
//   Stage: zero halo via ds_store, async-copy interior via
//          GLOBAL_LOAD_ASYNC_TO_LDS_B32, s_wait_asynccnt, barrier.
//   Wave:  for each 16-pixel group: A = 16x12 im2col taps from LDS,
//          3 chained V_WMMA_F32_16X16X4_F32; D cols 0/1 = gx/gy.
//   f32 A 16x4 layout (ISA 7.12.2): lanes 0-15 hold K0/K1 (M=lane),
//   lanes 16-31 hold K2/K3 (M=lane-16). B 4x16 analogous. D: VGPR r holds
//   rows r (lanes 0-15) and r+8 (lanes 16-31), N=lane%16.
// ---------------------------------------------------------------------------
__device__ __forceinline__ float tile_tap(const float (*tile)[kTileCols],
                                          int r, int x0, int mn, int t) {
    int row = (t < 9) ? (r + t / 3) : r;           // clamp keeps addr in-bounds
    int col = (t < 9) ? (x0 + mn + t % 3) : 0;
    float v = tile[row][col];
    return (t < 9) ? v : 0.0f;                     // pad taps 9..11
}

__global__ void __launch_bounds__(256)
k_sobel_wmma(const float* __restrict__ gray, const float* __restrict__ ew,
             float* __restrict__ grad, float* __restrict__ wgm) {
    __shared__ float tile[kTileRows][kTileCols];   // rows y0-1..y0+8, cols -1..224
    __shared__ float xch[8][32][8];
    __shared__ float sew[18];

    const int tid  = threadIdx.x;
    const int lane = tid & 31;
    const int wv   = tid >> 5;
    const int b    = blockIdx.x / (kH / kBand);    // 28 bands per image
    const int y0   = (blockIdx.x % (kH / kBand)) * kBand;
    const float* gb = gray + (size_t)b * kHW;

    if (tid < 18) sew[tid] = ew[tid];              // (2,1,3,3) flat

    // Zero the halo cells (disjoint from async-written interior: DS-pipe and
    // async-engine LDS writes are mutually unordered, so keep them disjoint).
    for (int i = tid; i < kTileRows * kTileCols; i += 256) {
        int r = i / kTileCols, cc = i % kTileCols;
        int gy = y0 + r - 1, gx = cc - 1;
        if (gy < 0 || gy >= kH || gx < 0 || gx >= kW) tile[r][cc] = 0.0f;
    }

    // Async-copy interior gray tile global -> LDS (ASYNCcnt path).
    const unsigned long long base = (unsigned long long)(uintptr_t)gb;
    for (int i = tid; i < kTileRows * kTileCols; i += 256) {
        int r = i / kTileCols, cc = i % kTileCols;
        int gy = y0 + r - 1, gx = cc - 1;
        if (gy >= 0 && gy < kH && gx >= 0 && gx < kW) {
            unsigned ldst = lds_addr_u32(&tile[r][cc]);
            unsigned voff = (unsigned)((gy * kW + gx) * 4);
            asm volatile("global_load_async_to_lds_b32 %0, %1, %2"
                         :: "v"(ldst), "v"(voff), "s"(base) : "memory");
        }
    }
    asm volatile("s_wait_asynccnt 0x0" ::: "memory");
    __syncthreads();

    const int mn   = lane & 15;                    // row M for A, col N for B
    const int kofs = (lane >= 16) ? 2 : 0;         // K sub-pair per lane half

    // B operand is x-group invariant: hoist out of the sweep.
    v2f bmat[3];
#pragma unroll
    for (int k = 0; k < 3; ++k) {
        const int t0 = 4 * k + kofs;
        bmat[k].x = (t0     < 9 && mn < 2) ? sew[mn * 9 + t0]     : 0.0f;
        bmat[k].y = (t0 + 1 < 9 && mn < 2) ? sew[mn * 9 + t0 + 1] : 0.0f;
    }

    const int r = wv;                              // wave <-> row in band
    const int y = y0 + r;
    for (int xg = 0; xg < kW / 16; ++xg) {
        const int x0 = xg * 16;
        v8f c = {0.f, 0.f, 0.f, 0.f, 0.f, 0.f, 0.f, 0.f};
#pragma unroll
        for (int k = 0; k < 3; ++k) {
            const int t0 = 4 * k + kofs;
            v2f a;
            a.x = tile_tap(tile, r, x0, mn, t0);
            a.y = tile_tap(tile, r, x0, mn, t0 + 1);
            c = __builtin_amdgcn_wmma_f32_16x16x4_f32(false, a, false, bmat[k],
                                                      (short)0, c, false, false);
        }
#pragma unroll
        for (int rr = 0; rr < 8; ++rr) xch[wv][lane][rr] = c[rr];
        // same-wave LDS ordering; no barrier needed
        if (lane < 16) {
            const int p = lane;
            float gx = xch[wv][p < 8 ? 0 : 16][p & 7];   // D[M=p][N=0]
            float gy = xch[wv][p < 8 ? 1 : 17][p & 7];   // D[M=p][N=1]
            float g2 = gx * gx + gy * gy + 1e-8f;
            int o = b * kHW + y * kW + x0 + p;
            grad[o] = sqrtf(g2);
            wgm[o]  = g2 * g2 * 10.0f;                    // grad^4 * GRAD_WEIGHT
        }
    }
}

// ---------------------------------------------------------------------------
// Sequential centroid minima claiming: one wave32 per image, LDS occupancy
// bitmap, wave-level reductions.
// ---------------------------------------------------------------------------
__global__ void k_minima(const float* __restrict__ grad, int* __restrict__ centsI,
                         float* __restrict__ centsF) {
    __shared__ unsigned occ[kHW / 32];               // 6272 bytes
    const int b = blockIdx.x, lane = threadIdx.x;
    for (int i = lane; i < kHW / 32; i += 32) occ[i] = 0u;
    __syncthreads();
    const float* g = grad + (size_t)b * kHW;

    for (int c = 0; c < kC; ++c) {
        const int cy = 16 * (c / 14) + 8, cx = 16 * (c % 14) + 8;
        const int y0 = max(0, cy - kNBnd), y1 = min(kH, cy + kNBnd);
        const int x0 = max(0, cx - kNBnd), x1 = min(kW, cx + kNBnd);
        const int ylen = y1 - y0, xlen = x1 - x0;

        float mv = __builtin_inff();
        for (int t = lane; t < kWin * kWin; t += 32) {
            int wy = t / kWin, wx = t - wy * kWin;
            if (wy < ylen && wx < xlen) mv = fminf(mv, g[(y0 + wy) * kW + x0 + wx]);
        }
        for (int m = 16; m > 0; m >>= 1) mv = fminf(mv, __shfl_xor(mv, m));

        int best = 0x7fffffff;                       // first (row-major) match
        for (int t = lane; t < kWin * kWin; t += 32) {
            int wy = t / kWin, wx = t - wy * kWin;
            if (wy < ylen && wx < xlen) {
                int pix = (y0 + wy) * kW + (x0 + wx);
                bool occd = (occ[pix >> 5] >> (pix & 31)) & 1u;
                if (!occd && g[pix] == mv) best = min(best, t);
            }
        }
        for (int m = 16; m > 0; m >>= 1) best = min(best, __shfl_xor(best, m));

        int ny = cy, nx = cx;
        const bool found = (best != 0x7fffffff);
        if (found) { ny = y0 + best / kWin; nx = x0 + best % kWin; }
        if (lane == 0) {
            if (found) { int pix = ny * kW + nx; occ[pix >> 5] |= 1u << (pix & 31); }
            int o = (b * kC + c) * 2;
            centsI[o] = ny; centsI[o + 1] = nx;
            centsF[o] = (float)ny; centsF[o + 1] = (float)nx;
        }
        __syncthreads();
    }
}

// ---------------------------------------------------------------------------
// Color diffs: only 2 unique maps (vertical, horizontal); the 4 directional
// cdiffs are shifted views of these.
// ---------------------------------------------------------------------------
__global__ void k_cdiff(const float* __restrict__ x, float* __restrict__ cdV,
                        float* __restrict__ cdH) {
    int i = blockIdx.x * blockDim.x + threadIdx.x;
    if (i >= kNPix) return;
    int b = i / kHW, p = i % kHW, iy = p / kW, ix = p % kW;
    int ydn = (iy + 1 == kH) ? 0 : iy + 1;
    int xrt = (ix + 1 == kW) ? 0 : ix + 1;
    const float* xb = x + (size_t)b * 3 * kHW;
    float sv = 0.f, sh = 0.f;
#pragma unroll
    for (int ch = 0; ch < 3; ++ch) {
        const float* xc = xb + ch * kHW;
        float v = xc[p];
        sv += fabsf(v - xc[ydn * kW + ix]);
        sh += fabsf(v - xc[iy * kW + xrt]);
    }
    cdV[i] = sv;
    cdH[i] = sh;
}

__global__ void k_init(float* __restrict__ dist, int* __restrict__ mask) {
    int i = blockIdx.x * blockDim.x + threadIdx.x;
    if (i >= kNPix) return;
    dist[i] = __builtin_inff();
    mask[i] = -1;
}

__global__ void k_seed(const int* __restrict__ centsI, float* __restrict__ dist,
                       int* __restrict__ mask) {
    int t = blockIdx.x * blockDim.x + threadIdx.x;
    if (t >= kB * kC) return;
    int b = t / kC, k = t % kC;
    int ny = centsI[2 * t], nx = centsI[2 * t + 1];
    int o = b * kHW + ny * kW + nx;
    dist[o] = 0.0f;
    mask[o] = k;
}

// ---------------------------------------------------------------------------
// One directional relaxation (ping-pong) == one inner step of the reference's
// sequential per-direction update. Working set (~90 MB) is L2-resident.
// ---------------------------------------------------------------------------
template <int DIR>
__global__ void k_prop(const float* __restrict__ din, const int* __restrict__ min_,
                       float* __restrict__ dout, int* __restrict__ mout,
                       const float* __restrict__ wgm, const float* __restrict__ cdV,
                       const float* __restrict__ cdH) {
    int i = blockIdx.x * blockDim.x + threadIdx.x;
    if (i >= kNPix) return;
    int b = i / kHW, p = i % kHW, iy = p / kW, ix = p % kW;
    const float* cV = cdV + (size_t)b * kHW;
    const float* cH = cdH + (size_t)b * kHW;
    int ni = iy, nj = ix;
    float cd;
    if (DIR == 0)      { ni = (iy + 1 == kH) ? 0 : iy + 1; cd = cV[p]; }
    else if (DIR == 1) { ni = (iy == 0) ? kH - 1 : iy - 1; cd = cV[ni * kW + ix]; }
    else if (DIR == 2) { nj = (ix + 1 == kW) ? 0 : ix + 1; cd = cH[p]; }
    else               { nj = (ix == 0) ? kW - 1 : ix - 1; cd = cH[iy * kW + nj]; }
    int nofs = b * kHW + ni * kW + nj;
    float wd = din[nofs] + wgm[i] + cd * kColorW;
    float dc = din[i];
    bool upd = wd < dc;
    dout[i] = upd ? wd : dc;
    mout[i] = upd ? min_[nofs] : min_[i];
}

__global__ void k_mask2f(const int* __restrict__ mask, float* __restrict__ out) {
    int i = blockIdx.x * blockDim.x + threadIdx.x;
    if (i >= kNPix) return;
    out[i] = (float)mask[i];
}

// ---------------------------------------------------------------------------
extern "C" void kernel_launch(void* const* d_in, const int* in_sizes, int n_in,
                              void* d_out, int out_size, void* d_ws, size_t ws_size,
                              hipStream_t stream) {
    const float* x  = (const float*)d_in[0];   // (64,3,224,224) f32
    const float* ew = (const float*)d_in[1];   // (2,1,3,3) Sobel f32

    float* out_grad  = (float*)d_out;                    // NPix
    float* out_cents = out_grad + kNPix;                 // 64*196*2
    float* out_mask  = out_cents + (size_t)kB * kC * 2;  // NPix
    float* gray      = out_mask;  // reuse mask output region as gray scratch

    float* ws    = (float*)d_ws;
    float* wgm   = ws;
    float* cdV   = ws + 1 * (size_t)kNPix;
    float* cdH   = ws + 2 * (size_t)kNPix;
    float* dist0 = ws + 3 * (size_t)kNPix;
    float* dist1 = ws + 4 * (size_t)kNPix;
    int*   mask0 = (int*)(ws + 5 * (size_t)kNPix);
    int*   mask1 = (int*)(ws + 6 * (size_t)kNPix);
    int*   centsI = (int*)(ws + 7 * (size_t)kNPix);

    const int TB = 256;
    const int gridPix = (kNPix + TB - 1) / TB;

    k_gray<<<gridPix, TB, 0, stream>>>(x, gray);
    k_sobel_wmma<<<kB * (kH / kBand), 256, 0, stream>>>(gray, ew, out_grad, wgm);
    k_minima<<<kB, 32, 0, stream>>>(out_grad, centsI, out_cents);
    k_cdiff<<<gridPix, TB, 0, stream>>>(x, cdV, cdH);
    k_init<<<gridPix, TB, 0, stream>>>(dist0, mask0);
    k_seed<<<(kB * kC + TB - 1) / TB, TB, 0, stream>>>(centsI, dist0, mask0);

    float* dc = dist0; float* dn = dist1;
    int*   mc = mask0; int*   mn = mask1;
    for (int it = 0; it < kIters; ++it) {
        k_prop<0><<<gridPix, TB, 0, stream>>>(dc, mc, dn, mn, wgm, cdV, cdH);
        { float* tf = dc; dc = dn; dn = tf; int* ti = mc; mc = mn; mn = ti; }
        k_prop<1><<<gridPix, TB, 0, stream>>>(dc, mc, dn, mn, wgm, cdV, cdH);
        { float* tf = dc; dc = dn; dn = tf; int* ti = mc; mc = mn; mn = ti; }
        k_prop<2><<<gridPix, TB, 0, stream>>>(dc, mc, dn, mn, wgm, cdV, cdH);
        { float* tf = dc; dc = dn; dn = tf; int* ti = mc; mc = mn; mn = ti; }
        k_prop<3><<<gridPix, TB, 0, stream>>>(dc, mc, dn, mn, wgm, cdV, cdH);
        { float* tf = dc; dc = dn; dn = tf; int* ti = mc; mc = mn; mn = ti; }
    }
    k_mask2f<<<gridPix, TB, 0, stream>>>(mc, out_mask);
}